// EFDM_loss_84482006713336
// MI455X (gfx1250) — compile-verified
//
#include <hip/hip_runtime.h>

// ---------------- problem constants (from reference setup_inputs) ----------------
// E group: M=2, B=8, C=256, H=64  -> HW=4096, ks=2048, kt=3072, CK = 256*3072 = 786432
// S group: M=2, B=8, C=512, H=32  -> HW=1024, ks=512,  kt=768,  CK = 512*768  = 393216
#define B_DIM 8
#define CK_E 786432
#define CK_S 393216
#define KPB  4096              // K per block in gram kernel
#define KW   (KPB / 8)         // K per wave (8 waves/block) = 512 -> 128 WMMA steps

// workspace layout (float offsets). Transposed tensors use a K-paired layout:
// element (m, k, b) lives at m*CK*8 + (k>>1)*16 + b*2 + (k&1), so one lane's
// two consecutive-K values are an aligned 8-byte pair (one global_load_b64).
#define OFF_SSE 0ull                                   // ss_E : 2*CK_E*8
#define OFF_STE (OFF_SSE + 2ull * CK_E * 8)            // st_E : 2*CK_E*8
#define OFF_SSS (OFF_STE + 2ull * CK_E * 8)            // ss_S : 2*CK_S*8
#define OFF_STS (OFF_SSS + 2ull * CK_S * 8)            // st_S : 2*CK_S*8
#define OFF_GE  (OFF_STS + 2ull * CK_S * 8)            // Gram E [m][16][16] : 512
#define OFF_GS  (OFF_GE + 512ull)                      // Gram S [m][16][16] : 512
#define OFF_SSSE (OFF_GS + 512ull)                     // sum ss^2 E [m][b]  : 16
#define OFF_STTE (OFF_SSSE + 16ull)                    // sum st^2 E [m][b]  : 16
#define OFF_SSSS (OFF_STTE + 16ull)                    // sum ss^2 S         : 16
#define OFF_STTS (OFF_SSSS + 16ull)                    // sum st^2 S         : 16
#define N_ACCUM 1088                                   // floats to zero (G's + sums)

typedef float v2f __attribute__((ext_vector_type(2)));
typedef float v8f __attribute__((ext_vector_type(8)));

__global__ void zero_accum_kernel(float* __restrict__ p, int n) {
    int i = blockIdx.x * blockDim.x + threadIdx.x;
    if (i < n) p[i] = 0.0f;
}

// ------------------------------------------------------------------
// Per-row masked bitonic sort in LDS, then select/interp and write in
// K-paired transposed layout for the WMMA kernel. Fuses the
// sum-of-squares reduction needed for the MSE expansion.
// One block (256 threads) per (m,b,c) row.
// ------------------------------------------------------------------
template <int N>
__global__ void sort_select_kernel(const float* __restrict__ x,
                                   const float* __restrict__ mask,
                                   float* __restrict__ outT,   // paired-transposed
                                   float* __restrict__ sumsq,  // [m][8]
                                   int C, int ks, int kt, int doInterp) {
    __shared__ float lds[N];
    __shared__ float red[256];
    const int tid = threadIdx.x;
    const int row = blockIdx.x;
    const int c = row % C;
    const int b = (row / C) % B_DIM;
    const int m = row / (C * B_DIM);
    const size_t base = (size_t)row * N;

    for (int i = tid; i < N; i += 256) {
        float v = x[base + i];
        float mv = mask[base + i];
        lds[i] = (mv > 0.5f) ? v : __builtin_inff();
    }
    __syncthreads();

    // bitonic sort ascending (masked-out values = +inf sink to the tail)
    for (int k = 2; k <= N; k <<= 1) {
        for (int j = k >> 1; j > 0; j >>= 1) {
            for (int i = tid; i < N; i += 256) {
                int ixj = i ^ j;
                if (ixj > i) {
                    float a = lds[i], bb = lds[ixj];
                    bool up = ((i & k) == 0);
                    if ((a > bb) == up) { lds[i] = bb; lds[ixj] = a; }
                }
            }
            __syncthreads();
        }
    }

    const size_t obase = (size_t)m * C * kt * B_DIM;
    float local = 0.0f;
    if (doInterp) {
        const float scale = (float)(ks - 1) / (float)(kt - 1);
        for (int t = tid; t < kt; t += 256) {
            float p = (float)t * scale;
            int i0 = (int)p;
            if (i0 > ks - 2) i0 = ks - 2;
            float fr = p - (float)i0;
            float v = lds[i0] * (1.0f - fr) + lds[i0 + 1] * fr;
            int gk = c * kt + t;
            outT[obase + (size_t)(gk >> 1) * 16 + (b << 1) + (gk & 1)] = v;
            local += v * v;
        }
    } else {
        for (int t = tid; t < kt; t += 256) {
            float v = lds[t];
            int gk = c * kt + t;
            outT[obase + (size_t)(gk >> 1) * 16 + (b << 1) + (gk & 1)] = v;
            local += v * v;
        }
    }
    red[tid] = local;
    __syncthreads();
    for (int s = 128; s > 0; s >>= 1) {
        if (tid < s) red[tid] += red[tid + s];
        __syncthreads();
    }
    if (tid == 0) atomicAdd(&sumsq[m * B_DIM + b], red[0]);
}

// ------------------------------------------------------------------
// G[m][bs][bt] = sum_k ssT[m][k][bs] * stT[m][k][bt] via chained
// V_WMMA_F32_16X16X4_F32. 256 threads = 8 waves; each wave owns a
// KW-wide K slice with a compile-time trip count (scalar loop control,
// EXEC stays all-ones around the WMMA — ISA 7.12 requirement).
// Pad lanes (row 8..15) alias rows 0..7: garbage lands only in the
// D rows/cols 8..15 of the tile, which finalize never reads.
// A layout (16x4 f32): lanes 0-15 -> K=0,1 ; lanes 16-31 -> K=2,3.
// One global_load_b64 per operand per WMMA thanks to the paired layout.
// ------------------------------------------------------------------
__global__ void gram_wmma_kernel(const float* __restrict__ ssT,
                                 const float* __restrict__ stT,
                                 float* __restrict__ G, // [m][16][16]
                                 int CK) {
    const int m = blockIdx.y;
    const int wave = threadIdx.x >> 5;
    const int lane = threadIdx.x & 31;
    const int row = lane & 15;
    const int half = lane >> 4;
    const int r8 = row & 7;

    const size_t mb = (size_t)m * CK * B_DIM;
    const int k0 = blockIdx.x * KPB + wave * KW;
    // lane's pair index at step k is (k>>1)+half ; address = pair*16 + r8*2
    const float* Aw = ssT + mb + ((size_t)(k0 >> 1) + half) * 16 + (r8 << 1);
    const float* Bw = stT + mb + ((size_t)(k0 >> 1) + half) * 16 + (r8 << 1);

    v8f acc = {};
#pragma unroll 8
    for (int it = 0; it < KW / 4; ++it) {
        // each WMMA step advances K by 4 -> pair index by 2 -> 32 floats
        v2f a = *(const v2f*)(Aw + (size_t)it * 32);
        v2f b = *(const v2f*)(Bw + (size_t)it * 32);
        acc = __builtin_amdgcn_wmma_f32_16x16x4_f32(
            /*neg_a=*/false, a, /*neg_b=*/false, b,
            /*c_mod=*/(short)0, acc, /*reuse_a=*/false, /*reuse_b=*/false);
    }

    // C/D layout: lane 0-15 -> N=lane, M=i ; lane 16-31 -> N=lane-16, M=i+8
    float* Gm = G + m * 256;
#pragma unroll
    for (int i = 0; i < 8; ++i) {
        int M = i + 8 * half;
        atomicAdd(&Gm[M * 16 + row], acc[i]);
    }
}

// ------------------------------------------------------------------
// loss = sum_b (pe+ps)/(ne+ns) with MSE expansion:
//   mse(bs,bt) = (Sss[bs] + Stt[bt] - 2*G[bs][bt]) / (C*kt)
// ------------------------------------------------------------------
__global__ void finalize_kernel(const float* __restrict__ GE, const float* __restrict__ GS,
                                const float* __restrict__ SssE, const float* __restrict__ SttE,
                                const float* __restrict__ SssS, const float* __restrict__ SttS,
                                const int* __restrict__ neg_idx, float* __restrict__ out) {
    if (threadIdx.x != 0 || blockIdx.x != 0) return;
    const float invE = 1.0f / (256.0f * 3072.0f);
    const float invS = 1.0f / (512.0f * 768.0f);
    float loss = 0.0f;
    for (int b = 0; b < B_DIM; ++b) {
        float pos = 0.0f, neg = 0.0f;
        for (int m = 0; m < 2; ++m) {
            pos += (SssE[m * 8 + b] + SttE[m * 8 + b] - 2.0f * GE[m * 256 + b * 16 + b]) * invE;
            pos += (SssS[m * 8 + b] + SttS[m * 8 + b] - 2.0f * GS[m * 256 + b * 16 + b]) * invS;
            for (int n = 0; n < 3; ++n) {
                int j = neg_idx[b * 3 + n];
                neg += (SssE[m * 8 + j] + SttE[m * 8 + b] - 2.0f * GE[m * 256 + j * 16 + b]) * invE;
                neg += (SssS[m * 8 + j] + SttS[m * 8 + b] - 2.0f * GS[m * 256 + j * 16 + b]) * invS;
            }
        }
        loss += pos / neg;
    }
    out[0] = loss;
}

extern "C" void kernel_launch(void* const* d_in, const int* in_sizes, int n_in,
                              void* d_out, int out_size, void* d_ws, size_t ws_size,
                              hipStream_t stream) {
    const float* style_E          = (const float*)d_in[0];
    const float* style_E_mask     = (const float*)d_in[1];
    const float* style_S          = (const float*)d_in[2];
    const float* style_S_mask     = (const float*)d_in[3];
    const float* translate_E      = (const float*)d_in[4];
    const float* translate_E_mask = (const float*)d_in[5];
    const float* translate_S      = (const float*)d_in[6];
    const float* translate_S_mask = (const float*)d_in[7];
    const int*   neg_idx          = (const int*)d_in[8];
    float* out = (float*)d_out;
    float* ws = (float*)d_ws;

    // zero Gram matrices + sum-of-squares accumulators
    zero_accum_kernel<<<(N_ACCUM + 255) / 256, 256, 0, stream>>>(ws + OFF_GE, N_ACCUM);

    // masked sort + select/interp + paired-transpose store + fused sum-of-squares
    // E: rows = 2*8*256 = 4096 of 4096 elems ; S: rows = 2*8*512 = 8192 of 1024 elems
    sort_select_kernel<4096><<<4096, 256, 0, stream>>>(
        style_E, style_E_mask, ws + OFF_SSE, ws + OFF_SSSE, 256, 2048, 3072, 1);
    sort_select_kernel<4096><<<4096, 256, 0, stream>>>(
        translate_E, translate_E_mask, ws + OFF_STE, ws + OFF_STTE, 256, 3072, 3072, 0);
    sort_select_kernel<1024><<<8192, 256, 0, stream>>>(
        style_S, style_S_mask, ws + OFF_SSS, ws + OFF_SSSS, 512, 512, 768, 1);
    sort_select_kernel<1024><<<8192, 256, 0, stream>>>(
        translate_S, translate_S_mask, ws + OFF_STS, ws + OFF_STTS, 512, 768, 768, 0);

    // fp32 WMMA Gram matrices over K = C*kt
    gram_wmma_kernel<<<dim3(CK_E / KPB, 2), 256, 0, stream>>>(
        ws + OFF_SSE, ws + OFF_STE, ws + OFF_GE, CK_E);
    gram_wmma_kernel<<<dim3(CK_S / KPB, 2), 256, 0, stream>>>(
        ws + OFF_SSS, ws + OFF_STS, ws + OFF_GS, CK_S);

    finalize_kernel<<<1, 32, 0, stream>>>(
        ws + OFF_GE, ws + OFF_GS, ws + OFF_SSSE, ws + OFF_STTE,
        ws + OFF_SSSS, ws + OFF_STTS, neg_idx, out);
}